// Model_55473797595459
// MI455X (gfx1250) — compile-verified
//
#include <hip/hip_runtime.h>
#include <math.h>

typedef __attribute__((ext_vector_type(2))) float v2f;
typedef __attribute__((ext_vector_type(8))) float v8f;

#define B_SZ 4
#define N_NODES 10000
#define BN (B_SZ * N_NODES)
#define LN_EPS 1e-5f

// ---------------------------------------------------------------------------
// Tiled GEMM: C[rows x NOUT] = act(A[rows x K] @ W[K x NOUT] + bias)
// Compile-time K/NOUT -> constant-stride loads; each wave computes a 64x16
// output slab (4 row-tiles sharing one B panel -> 4 WMMAs per B load).
// ACT: 0 = none, 1 = relu, 2 = gelu(exact)
// ---------------------------------------------------------------------------
template <int ACT>
__device__ inline float apply_act(float v)
{
    if (ACT == 1) return fmaxf(v, 0.f);
    if (ACT == 2) return 0.5f * v * (1.f + erff(v * 0.70710678118654752f));
    return v;
}

template <int K, int NOUT, int ACT>
__global__ __launch_bounds__(256)
void gemm_wmma(const float* __restrict__ A, const float* __restrict__ W,
               const float* __restrict__ bias, float* __restrict__ C, int rows)
{
    const int w    = threadIdx.x >> 5;
    const int lane = threadIdx.x & 31;
    const int nct  = NOUT >> 4;                               // col tiles
    const long total = (long)(rows >> 6) * nct;               // 64-row super-tiles
    const long st = (long)blockIdx.x * 8 + w;
    if (st >= total) return;                                  // wave-uniform exit
    const int ct   = (int)(st % nct);
    const int rt64 = (int)(st / nct);

    const int lm = lane & 15;
    const int hi = lane >> 4;

    // Per-lane A rows for the 4 stacked 16-row tiles
    const float* A0 = A + (size_t)(rt64 * 64 + lm) * K;
    const float* A1 = A0 + (size_t)16 * K;
    const float* A2 = A0 + (size_t)32 * K;
    const float* A3 = A0 + (size_t)48 * K;
    const float* Wc = W + ct * 16 + lm;                       // this lane's W column

    v8f c0 = {}, c1 = {}, c2 = {}, c3 = {};
    for (int k = 0; k < K; k += 4) {
        const int ka = k + 2 * hi;                            // lanes 0-15: k,k+1 ; 16-31: k+2,k+3
        v2f b;
        b.x = Wc[(size_t)ka * NOUT];
        b.y = Wc[(size_t)(ka + 1) * NOUT];
        const v2f a0 = *(const v2f*)(A0 + ka);
        const v2f a1 = *(const v2f*)(A1 + ka);
        const v2f a2 = *(const v2f*)(A2 + ka);
        const v2f a3 = *(const v2f*)(A3 + ka);
        c0 = __builtin_amdgcn_wmma_f32_16x16x4_f32(false, a0, false, b, (short)0, c0, false, false);
        c1 = __builtin_amdgcn_wmma_f32_16x16x4_f32(false, a1, false, b, (short)0, c1, false, false);
        c2 = __builtin_amdgcn_wmma_f32_16x16x4_f32(false, a2, false, b, (short)0, c2, false, false);
        c3 = __builtin_amdgcn_wmma_f32_16x16x4_f32(false, a3, false, b, (short)0, c3, false, false);
    }

    const int col  = ct * 16 + lm;
    const float bv = bias ? bias[col] : 0.f;
    float* Cp = C + (size_t)(rt64 * 64) * NOUT + col;
    #pragma unroll
    for (int v = 0; v < 8; ++v) {
        const int r = v + 8 * hi;                             // D layout: vgpr v -> M=v / v+8
        Cp[(size_t)(r +  0) * NOUT] = apply_act<ACT>(c0[v] + bv);
        Cp[(size_t)(r + 16) * NOUT] = apply_act<ACT>(c1[v] + bv);
        Cp[(size_t)(r + 32) * NOUT] = apply_act<ACT>(c2[v] + bv);
        Cp[(size_t)(r + 48) * NOUT] = apply_act<ACT>(c3[v] + bv);
    }
}

template <int K, int NOUT, int ACT>
static void launch_gemm(const float* A, const float* W, const float* bias, float* C,
                        int rows, hipStream_t st)
{
    long supertiles = (long)(rows / 64) * (NOUT / 16);
    int blocks = (int)((supertiles + 7) / 8);
    gemm_wmma<K, NOUT, ACT><<<blocks, 256, 0, st>>>(A, W, bias, C, rows);
}

// ---------------------------------------------------------------------------
// agg[b,n,:256] = bias[:]   (feature width fixed at 256 for GCN layers)
// ---------------------------------------------------------------------------
__global__ void fill_bias256(float* __restrict__ out, const float* __restrict__ bias, long total)
{
    long i = (long)blockIdx.x * 256 + threadIdx.x;
    if (i < total) out[i] = bias[i & 255];
}

// ---------------------------------------------------------------------------
// Edge scatter: agg[b, dst[e], f] += sup[b, src[e], f], 256 features, float4/thread.
// 64 consecutive threads cover one edge row -> coalesced; atomics are L2-resident.
// ---------------------------------------------------------------------------
__global__ void scatter_edges(const float* __restrict__ sup, float* __restrict__ agg,
                              const int* __restrict__ src, const int* __restrict__ dst, int E)
{
    long idx = (long)blockIdx.x * 256 + threadIdx.x;
    long total = (long)B_SZ * E * 64;
    if (idx >= total) return;
    int  fc = (int)(idx & 63);
    long t  = idx >> 6;
    int  e  = (int)(t % E);
    int  b  = (int)(t / E);
    int  s  = src[e];
    int  d  = dst[e];
    const float4 v = *(const float4*)(sup + ((size_t)b * N_NODES + s) * 256 + fc * 4);
    float* p = agg + ((size_t)b * N_NODES + d) * 256 + fc * 4;
    unsafeAtomicAdd(p + 0, v.x);
    unsafeAtomicAdd(p + 1, v.y);
    unsafeAtomicAdd(p + 2, v.z);
    unsafeAtomicAdd(p + 3, v.w);
}

__global__ void gelu_k(float* __restrict__ x, long total)
{
    long i = (long)blockIdx.x * 256 + threadIdx.x;
    if (i < total) {
        float v = x[i];
        x[i] = 0.5f * v * (1.f + erff(v * 0.70710678118654752f));
    }
}

// ---------------------------------------------------------------------------
// a[b,n] = tanh(x[b,n,:] @ W(256x8)) . u(8)    -- one wave per row
// ---------------------------------------------------------------------------
__global__ void attn_score(const float* __restrict__ x, const float* __restrict__ W,
                           const float* __restrict__ u, float* __restrict__ a)
{
    const int wv = (int)((blockIdx.x * (long)blockDim.x + threadIdx.x) >> 5);
    const int ln = threadIdx.x & 31;
    if (wv >= BN) return;
    const float* row = x + (size_t)wv * 256;
    float acc[8] = {0.f, 0.f, 0.f, 0.f, 0.f, 0.f, 0.f, 0.f};
    for (int k = ln; k < 256; k += 32) {
        const float xv = row[k];
        const float* wr = W + k * 8;
        #pragma unroll
        for (int j = 0; j < 8; ++j) acc[j] += xv * wr[j];
    }
    #pragma unroll
    for (int j = 0; j < 8; ++j)
        for (int off = 16; off > 0; off >>= 1)
            acc[j] += __shfl_xor(acc[j], off, 32);
    if (ln == 0) {
        float s = 0.f;
        #pragma unroll
        for (int j = 0; j < 8; ++j) s += tanhf(acc[j]) * u[j];
        a[wv] = s;
    }
}

// ---------------------------------------------------------------------------
// Attention pooling with 3 refinement iterations. One block (1024 thr) per batch.
// ---------------------------------------------------------------------------
__global__ __launch_bounds__(1024)
void attn_pool(const float* __restrict__ x, float* __restrict__ sc,
               float* __restrict__ r_out)
{
    const int b = blockIdx.x;
    const int t = threadIdx.x;
    const float* xb = x + (size_t)b * N_NODES * 256;
    float* ab = sc + (size_t)b * N_NODES;

    __shared__ float red[1024];
    __shared__ float rsh[256];
    __shared__ float stat;

    const int d  = t & 255, g  = t >> 8;   // 4 n-strides x 256 features
    const int wv = t >> 5,  ln = t & 31;   // 32 waves

    for (int it = 0; it < 4; ++it) {
        if (it > 0) {
            // s[n] = x[b,n,:] . r      (one wave per node, coalesced over d)
            for (int n = wv; n < N_NODES; n += 32) {
                float acc = 0.f;
                for (int dd = ln; dd < 256; dd += 32)
                    acc += xb[(size_t)n * 256 + dd] * rsh[dd];
                for (int off = 16; off > 0; off >>= 1)
                    acc += __shfl_xor(acc, off, 32);
                if (ln == 0) ab[n] = acc;
            }
            __syncthreads();
        }
        // ---- max over scores ----
        float m = -3.4e38f;
        for (int n = t; n < N_NODES; n += 1024) m = fmaxf(m, ab[n]);
        red[t] = m; __syncthreads();
        for (int off = 512; off > 0; off >>= 1) {
            if (t < off) red[t] = fmaxf(red[t], red[t + off]);
            __syncthreads();
        }
        if (t == 0) stat = red[0];
        __syncthreads();
        const float mx = stat;
        // ---- sum of exp ----
        float s = 0.f;
        for (int n = t; n < N_NODES; n += 1024) s += expf(ab[n] - mx);
        red[t] = s; __syncthreads();
        for (int off = 512; off > 0; off >>= 1) {
            if (t < off) red[t] += red[t + off];
            __syncthreads();
        }
        if (t == 0) stat = red[0];
        __syncthreads();
        const float inv = 1.f / stat;
        __syncthreads();
        // ---- r[d] = sum_n softmax(s)[n] * x[b,n,d] ----
        float p = 0.f;
        for (int n = g; n < N_NODES; n += 4)
            p += expf(ab[n] - mx) * xb[(size_t)n * 256 + d];
        red[t] = p * inv; __syncthreads();
        if (t < 256) rsh[t] = red[t] + red[t + 256] + red[t + 512] + red[t + 768];
        __syncthreads();
    }
    if (t < 256) r_out[b * 256 + t] = rsh[t];
}

// ---------------------------------------------------------------------------
// x[row,:] = LayerNorm(x[row,:] + other)   in place; one 256-thread block/row.
// bcast=1: other indexed [b*256+t] (pooled r); bcast=0: other[row*256+t] (residual h)
// ---------------------------------------------------------------------------
__global__ __launch_bounds__(256)
void add_ln(float* __restrict__ x, const float* __restrict__ other, int bcast,
            const float* __restrict__ g, const float* __restrict__ be)
{
    const int row = blockIdx.x;
    const int t   = threadIdx.x;
    const int b   = row / N_NODES;
    const size_t xi = (size_t)row * 256 + t;
    float v = x[xi] + (bcast ? other[b * 256 + t] : other[xi]);

    __shared__ float red[256];
    __shared__ float mu_s, var_s;
    red[t] = v; __syncthreads();
    for (int off = 128; off > 0; off >>= 1) {
        if (t < off) red[t] += red[t + off];
        __syncthreads();
    }
    if (t == 0) mu_s = red[0] * (1.f / 256.f);
    __syncthreads();
    const float dv = v - mu_s;
    red[t] = dv * dv; __syncthreads();
    for (int off = 128; off > 0; off >>= 1) {
        if (t < off) red[t] += red[t + off];
        __syncthreads();
    }
    if (t == 0) var_s = red[0] * (1.f / 256.f);
    __syncthreads();
    x[xi] = dv * rsqrtf(var_s + LN_EPS) * g[t] + be[t];
}

// ---------------------------------------------------------------------------
// Final head: mean_n(x @ fcW + fcb) == (mean_n x) @ fcW + fcb
// ---------------------------------------------------------------------------
__global__ __launch_bounds__(256)
void colsum(const float* __restrict__ x, float* __restrict__ cs)
{
    const int b = blockIdx.x;
    const int t = threadIdx.x;
    float s = 0.f;
    for (int n = 0; n < N_NODES; ++n)
        s += x[((size_t)b * N_NODES + n) * 256 + t];
    cs[b * 256 + t] = s;
}

__global__ void final_fc(const float* __restrict__ cs, const float* __restrict__ fcW,
                         const float* __restrict__ fcb, float* __restrict__ out)
{
    const int i = threadIdx.x;
    if (i >= B_SZ * 10) return;
    const int b = i / 10, o = i % 10;
    float s = 0.f;
    for (int dd = 0; dd < 256; ++dd)
        s += cs[b * 256 + dd] * fcW[dd * 10 + o];
    out[i] = s * (1.f / (float)N_NODES) + fcb[o];
}

// ---------------------------------------------------------------------------
// Host-side orchestration
// ---------------------------------------------------------------------------
extern "C" void kernel_launch(void* const* d_in, const int* in_sizes, int n_in,
                              void* d_out, int out_size, void* d_ws, size_t ws_size,
                              hipStream_t stream)
{
    const float* x_in = (const float*)d_in[0];
    const int*   src  = (const int*)  d_in[1];
    const int*   dst  = (const int*)  d_in[2];
    const float* Wc0  = (const float*)d_in[3];
    const float* bc0  = (const float*)d_in[4];
    const float* Wcs  = (const float*)d_in[5];
    const float* bcs  = (const float*)d_in[6];
    const float* mW1  = (const float*)d_in[7];
    const float* mb1  = (const float*)d_in[8];
    const float* mW2  = (const float*)d_in[9];
    const float* mb2  = (const float*)d_in[10];
    const float* mW3  = (const float*)d_in[11];
    const float* mb3  = (const float*)d_in[12];
    const float* ln1g = (const float*)d_in[13];
    const float* ln1b = (const float*)d_in[14];
    const float* ln2g = (const float*)d_in[15];
    const float* ln2b = (const float*)d_in[16];
    const float* attW = (const float*)d_in[17];
    const float* attu = (const float*)d_in[18];
    const float* fW1  = (const float*)d_in[19];
    const float* fb1  = (const float*)d_in[20];
    const float* fW2  = (const float*)d_in[21];
    const float* fb2  = (const float*)d_in[22];
    const float* fcW  = (const float*)d_in[23];
    const float* fcb  = (const float*)d_in[24];
    float* out = (float*)d_out;

    const int E = in_sizes[1];

    // workspace layout (floats), ~123 MB total
    const size_t BUF = (size_t)BN * 256;
    float* X  = (float*)d_ws;
    float* T1 = X  + BUF;
    float* T2 = T1 + BUF;
    float* SC = T2 + BUF;                         // BN scores
    float* R  = SC + BN;                          // B*256 pooled
    float* CS = R  + B_SZ * 256;                  // B*256 column sums

    const long elems = (long)BN * 256;
    const int  eb    = (int)((elems + 255) / 256);
    const long etot  = (long)B_SZ * E * 64;
    const int  sb    = (int)((etot + 255) / 256);

    // ---- GCN stack (4 layers) ----
    launch_gemm<128, 256, 0>(x_in, Wc0, nullptr, T1, BN, stream);
    fill_bias256<<<eb, 256, 0, stream>>>(T2, bc0, elems);
    scatter_edges<<<sb, 256, 0, stream>>>(T1, T2, src, dst, E);
    gelu_k<<<eb, 256, 0, stream>>>(T2, elems);
    float* cur = T2;  float* oth = X;
    for (int i = 0; i < 3; ++i) {
        launch_gemm<256, 256, 0>(cur, Wcs + (size_t)i * 256 * 256, nullptr, T1, BN, stream);
        fill_bias256<<<eb, 256, 0, stream>>>(oth, bcs + i * 256, elems);
        scatter_edges<<<sb, 256, 0, stream>>>(T1, oth, src, dst, E);
        gelu_k<<<eb, 256, 0, stream>>>(oth, elems);
        float* tmp = cur; cur = oth; oth = tmp;
    }
    // cur = X, oth = T2

    // ---- MLP: 256 -> 128 -> 64 -> 256 ----
    launch_gemm<256, 128, 1>(cur, mW1, mb1, T1, BN, stream);
    launch_gemm<128,  64, 1>(T1,  mW2, mb2, oth, BN, stream);
    launch_gemm< 64, 256, 0>(oth, mW3, mb3, T1, BN, stream);
    cur = T1;                                     // X and T2 now free

    // ---- 2 attention-pool + FFN blocks ----
    for (int i = 0; i < 2; ++i) {
        attn_score<<<(BN * 32 + 255) / 256, 256, 0, stream>>>(
            cur, attW + (size_t)i * 256 * 8, attu + i * 8, SC);
        attn_pool<<<B_SZ, 1024, 0, stream>>>(cur, SC, R);
        add_ln<<<BN, 256, 0, stream>>>(cur, R, 1, ln1g + i * 256, ln1b + i * 256);
        launch_gemm<256, 256, 1>(cur, fW1 + (size_t)i * 256 * 256, fb1 + i * 256, T2, BN, stream);
        launch_gemm<256, 256, 0>(T2,  fW2 + (size_t)i * 256 * 256, fb2 + i * 256, X,  BN, stream);
        add_ln<<<BN, 256, 0, stream>>>(cur, X, 0, ln2g + i * 256, ln2b + i * 256);
    }

    // ---- final head ----
    colsum<<<B_SZ, 256, 0, stream>>>(cur, CS);
    final_fc<<<1, 64, 0, stream>>>(CS, fcW, fcb, out);
}